// reblurWithKernel_70858370449556
// MI455X (gfx1250) — compile-verified
//
#include <hip/hip_runtime.h>
#include <stdint.h>

#define HH 256
#define WW 256
#define KS 33
#define NK (KS * KS)          // 1089 = 33 rows x 33 cols, no tail
#define PADR 16
#define TX 16
#define TY 16
#define TW (TX + 2 * PADR)    // 48
#define TH_ (TY + 2 * PADR)   // 48
#define KCHUNK KS             // one 33-tap kernel row per chunk
#define NCHUNK KS             // 33 chunks
#define THRESH 1.0e-4f
#define PLANE (HH * WW)       // 65536 elements per k-plane

// Low 32 bits of a flat (generic) LDS address are the LDS byte offset
// (shared aperture lives entirely in ADDR[63:32] on CDNA5), which is what
// the async-to-LDS VDST operand expects.
__device__ __forceinline__ uint32_t lds_off(const void* p) {
  return (uint32_t)(uintptr_t)p;
}

__global__ __launch_bounds__(256) void reblur_dynconv(
    const float* __restrict__ img,
    const float* __restrict__ ker,
    float* __restrict__ out)
{
  __shared__ float4 simg[TH_ * TW];           // 36,864 B: padded tile, 3ch packed
  __shared__ float  swt[2][KCHUNK * 256];     // 67,584 B: double-buffered weight rows

  const int tid = threadIdx.x;
  const int px = tid & 15, py = tid >> 4;
  const int x0 = (blockIdx.x & 15) * TX;
  const int y0 = (blockIdx.x >> 4) * TY;

  // ---- Stage replication-padded 48x48 image tile (edge clamp) ----
  for (int i = tid; i < TH_ * TW; i += 256) {
    int iy = i / TW, ix = i - iy * TW;
    int gy = min(max(y0 + iy - PADR, 0), HH - 1);
    int gx = min(max(x0 + ix - PADR, 0), WW - 1);
    int g = gy * WW + gx;
    simg[i] = make_float4(img[g], img[g + PLANE], img[g + 2 * PLANE], 0.f);
  }
  __syncthreads();

  // GVS addressing: uniform 64-bit base in SGPRs, per-thread 32-bit byte
  // offset in one VGPR (max offset ~285MB < 2^31).
  const uint64_t kbase   = (uint64_t)ker;
  const uint32_t pixoff4 = (uint32_t)((y0 + py) * WW + (x0 + px)) * 4u;

  // ---- Async stream of one 33-tap kernel row per chunk (non-temporal) ----
  auto issue_chunk = [&](int c) {
    const int buf = c & 1;
    const uint32_t chunkoff = pixoff4 + (uint32_t)(c * KS) * (uint32_t)(PLANE * 4);
    #pragma unroll
    for (int kc = 0; kc < KCHUNK; ++kc) {
      uint32_t voff = chunkoff + (uint32_t)kc * (uint32_t)(PLANE * 4);
      uint32_t la   = lds_off(&swt[buf][kc * 256 + tid]);
      asm volatile("global_load_async_to_lds_b32 %0, %1, %2 th:TH_LOAD_NT"
                   :: "v"(la), "v"(voff), "s"(kbase) : "memory");
    }
  };

  issue_chunk(0);
  issue_chunk(1);

  float ax = 0.f, ay = 0.f, az = 0.f;

  for (int c = 0; c < NCHUNK; ++c) {
    // In-order ASYNCcnt: <=33 outstanding means chunk c has landed in LDS.
    if (c + 1 < NCHUNK)
      asm volatile("s_wait_asynccnt 33" ::: "memory");
    else
      asm volatile("s_wait_asynccnt 0" ::: "memory");

    // dy == c (wave-uniform); dx == kc (compile-time after unroll):
    // every LDS access below is base-register + immediate offset.
    const float*  wrow = &swt[c & 1][tid];
    const float4* prow = &simg[(py + c) * TW + px];
    #pragma unroll
    for (int kc = 0; kc < KCHUNK; ++kc) {
      float w = wrow[kc * 256];
      w = (w > THRESH) ? w : 0.f;     // reference's tap-skip threshold
      float4 p = prow[kc];
      ax = fmaf(w, p.x, ax);
      ay = fmaf(w, p.y, ay);
      az = fmaf(w, p.z, az);
    }
    if (c + 2 < NCHUNK) issue_chunk(c + 2);   // refill the buffer just drained
  }

  const int o = (y0 + py) * WW + (x0 + px);
  out[o]             = ax;
  out[o + PLANE]     = ay;
  out[o + 2 * PLANE] = az;
}

extern "C" void kernel_launch(void* const* d_in, const int* in_sizes, int n_in,
                              void* d_out, int out_size, void* d_ws, size_t ws_size,
                              hipStream_t stream) {
  const float* img = (const float*)d_in[0];   // [1,3,256,256] f32
  const float* ker = (const float*)d_in[1];   // [1,1089,256,256] f32
  float* out = (float*)d_out;                 // [1,3,256,256] f32
  (void)in_sizes; (void)n_in; (void)out_size; (void)d_ws; (void)ws_size;
  reblur_dynconv<<<dim3(256), dim3(256), 0, stream>>>(img, ker, out);
}